// GraphSAGEModel_2843268350707
// MI455X (gfx1250) — compile-verified
//
#include <hip/hip_runtime.h>
#include <stdint.h>

#define N_NODES  300000
#define N_EDGES  600000
#define N_GRAPHS 512

typedef __attribute__((ext_vector_type(2))) float v2f;
typedef __attribute__((ext_vector_type(8))) float v8f;

// ---------------- utility ----------------
__global__ void k_zero(float* __restrict__ p, long n) {
  long i = (long)blockIdx.x * blockDim.x + threadIdx.x;
  long stride = (long)gridDim.x * blockDim.x;
  for (; i < n; i += stride) p[i] = 0.0f;
}

__global__ void k_deg(const long long* __restrict__ dst, float* __restrict__ deg, int E) {
  int e = blockIdx.x * blockDim.x + threadIdx.x;
  if (e < E) atomicAdd(&deg[(size_t)dst[e]], 1.0f);
}

__global__ void k_invdeg(float* __restrict__ d, int n) {
  int i = blockIdx.x * blockDim.x + threadIdx.x;
  if (i < n) d[i] = 1.0f / fmaxf(d[i], 1.0f);
}

// zero-pad weights: W[dout][din] -> Wp[dout][lstride], lstride = pow2 >= din
__global__ void k_padw(const float* __restrict__ W, float* __restrict__ Wp,
                       int din, int dout, int lstride, int lshift) {
  int i = blockIdx.x * blockDim.x + threadIdx.x;
  if (i < (dout << lshift)) {
    int n = i >> lshift;
    int k = i & (lstride - 1);
    Wp[i] = (k < din) ? W[n * din + k] : 0.0f;
  }
}

// ---------------- aggregation (scatter-add) ----------------
__global__ void k_scatter(const float* __restrict__ H, const long long* __restrict__ src,
                          const long long* __restrict__ dst, float* __restrict__ Agg,
                          int E, int d, int cshift) {
  long t = (long)blockIdx.x * blockDim.x + threadIdx.x;
  int e = (int)(t >> cshift);
  int c = (int)(t & ((1 << cshift) - 1));
  if (e < E && c < d) {
    size_t s  = (size_t)src[e];
    size_t dd = (size_t)dst[e];
    atomicAdd(&Agg[dd * (size_t)d + c], H[s * (size_t)d + c]);
  }
}

__global__ void k_scatter128(const float* __restrict__ H, const long long* __restrict__ src,
                             const long long* __restrict__ dst, float* __restrict__ Agg, int E) {
  long t = (long)blockIdx.x * blockDim.x + threadIdx.x;
  int e  = (int)(t >> 5);
  int c4 = (int)(t & 31) << 2;
  if (e < E) {
    size_t s  = (size_t)src[e];
    size_t dd = (size_t)dst[e];
    const float4 v = *(const float4*)(H + s * 128 + c4);
    float* o = Agg + dd * 128 + c4;
    atomicAdd(o + 0, v.x); atomicAdd(o + 1, v.y);
    atomicAdd(o + 2, v.z); atomicAdd(o + 3, v.w);
  }
}

// ---------------- SAGE GEMM with f32 WMMA ----------------
// Out[16-row tile] = (Agg*invdeg) @ Wl^T + X @ Wr^T + bias
// Weights pre-padded to [dout][lstride] with zeros -> branch-free inner loop.
// block = 32 * (dout/16) threads; one wave per 16-col output tile.
__global__ void k_sage_gemm(const float* __restrict__ Agg, const float* __restrict__ X,
                            const float* __restrict__ invdeg,
                            const float* __restrict__ Wlp, const float* __restrict__ Wrp,
                            const float* __restrict__ bias, float* __restrict__ Out,
                            int din, int dout, int lstride, int lshift) {
  extern __shared__ float lds[];
  float* ldsA = lds;                       // agg * invdeg
  float* ldsX = lds + (16 << lshift);      // root features
  const int r0  = blockIdx.x << 4;
  const int tid = threadIdx.x;

  if (din == lstride) {
    // din is pow2 (128): vectorized float4 staging
    const int chunks = (16 << lshift) >> 2;
    const int cshift = lshift - 2;
    for (int i = tid; i < chunks; i += blockDim.x) {
      int row = i >> cshift;
      int k4  = (i & ((1 << cshift) - 1)) << 2;
      size_t gi = (size_t)(r0 + row) * (size_t)din + k4;
      float4 av = *(const float4*)(Agg + gi);
      float4 xv = *(const float4*)(X + gi);
      float s = invdeg[r0 + row];
      av.x *= s; av.y *= s; av.z *= s; av.w *= s;
      *(float4*)(ldsA + (row << lshift) + k4) = av;
      *(float4*)(ldsX + (row << lshift) + k4) = xv;
    }
  } else {
    // generic scalar staging with zero padding (layer 1: din=18)
    const int tot = 16 << lshift;
    for (int i = tid; i < tot; i += blockDim.x) {
      int row = i >> lshift;
      int k   = i & (lstride - 1);
      float a = 0.0f, x = 0.0f;
      if (k < din) {
        size_t gi = (size_t)(r0 + row) * (size_t)din + k;
        a = Agg[gi] * invdeg[r0 + row];
        x = X[gi];
      }
      ldsA[i] = a;
      ldsX[i] = x;
    }
  }
  __syncthreads();

  const int lane = tid & 31;
  const int m    = lane & 15;
  const int n0   = (tid >> 5) << 4;
  const int kh   = (lane >> 4) << 1;    // 0 or 2 (K sub-offset per lane half)
  const int ncol = n0 + m;
  const int din4 = (din + 3) & ~3;

  v8f acc;
  const float bv = bias[ncol];
#pragma unroll
  for (int v = 0; v < 8; ++v) acc[v] = bv;

  const float* ldA = ldsA + (m << lshift);
  const float* ldX = ldsX + (m << lshift);
  const float* wl  = Wlp + (ncol << lshift);
  const float* wr  = Wrp + (ncol << lshift);

#pragma unroll 4
  for (int k0 = 0; k0 < din4; k0 += 4) {
    const int ka = k0 + kh;                         // 8-byte aligned offsets
    v2f a  = *(const v2f*)(ldA + ka);               // ds_load_b64
    v2f x  = *(const v2f*)(ldX + ka);
    v2f bl = *(const v2f*)(wl + ka);                // global_load_b64 (zero-padded)
    v2f br = *(const v2f*)(wr + ka);
    acc = __builtin_amdgcn_wmma_f32_16x16x4_f32(false, a, false, bl, (short)0, acc, false, false);
    acc = __builtin_amdgcn_wmma_f32_16x16x4_f32(false, x, false, br, (short)0, acc, false, false);
  }

  const int mb = (lane >> 4) << 3;      // rows 0..7 or 8..15
#pragma unroll
  for (int v = 0; v < 8; ++v) {
    Out[(size_t)(r0 + mb + v) * dout + ncol] = acc[v];
  }
}

// ---------------- BatchNorm ----------------
__global__ void k_colstats(const float* __restrict__ P, int nrows, int d, int rpb,
                           float* __restrict__ sum, float* __restrict__ sumsq) {
  const int c = threadIdx.x;                      // blockDim.x == d
  long r0 = (long)blockIdx.x * rpb;
  long r1 = r0 + rpb; if (r1 > nrows) r1 = nrows;
  float s = 0.0f, s2 = 0.0f;
  for (long r = r0; r < r1; ++r) {
    float v = P[r * d + c];
    s += v; s2 += v * v;
  }
  atomicAdd(&sum[c], s);
  atomicAdd(&sumsq[c], s2);
}

__global__ void k_bnfinalize(const float* __restrict__ sum, const float* __restrict__ sumsq,
                             const float* __restrict__ g, const float* __restrict__ be,
                             float* __restrict__ scale, float* __restrict__ shift,
                             int d, float invN) {
  int c = threadIdx.x;
  if (c < d) {
    float mu  = sum[c] * invN;
    float var = sumsq[c] * invN - mu * mu;
    float sc  = g[c] * rsqrtf(var + 1e-5f);
    scale[c] = sc;
    shift[c] = be[c] - mu * sc;
  }
}

__global__ void k_bnrelu(float* __restrict__ P, const float* __restrict__ scale,
                         const float* __restrict__ shift, long n, int dmask) {
  long i = (long)blockIdx.x * blockDim.x + threadIdx.x;
  if (i < n) {
    int c = (int)(i & dmask);
    P[i] = fmaxf(fmaf(P[i], scale[c], shift[c]), 0.0f);
  }
}

// ---------------- pooling + head ----------------
__global__ void k_pool(const float* __restrict__ H, const long long* __restrict__ batch,
                       float* __restrict__ psum, float* __restrict__ pcnt, int nrows) {
  long t = (long)blockIdx.x * blockDim.x + threadIdx.x;
  int nidx = (int)(t >> 6);
  int c    = (int)(t & 63);
  if (nidx < nrows) {
    int g = (int)batch[nidx];
    atomicAdd(&psum[(size_t)g * 64 + c], H[(size_t)nidx * 64 + c]);
    if (c == 0) atomicAdd(&pcnt[g], 1.0f);
  }
}

__global__ void k_head(const float* __restrict__ psum, const float* __restrict__ pcnt,
                       const float* __restrict__ w1, const float* __restrict__ b1,
                       const float* __restrict__ w2, const float* __restrict__ b2,
                       float* __restrict__ out) {
  int g = blockIdx.x * blockDim.x + threadIdx.x;
  if (g >= N_GRAPHS) return;
  float inv = 1.0f / fmaxf(pcnt[g], 1.0f);
  float p[64];
#pragma unroll
  for (int c = 0; c < 64; ++c) p[c] = psum[(size_t)g * 64 + c] * inv;
  float z[32];
#pragma unroll
  for (int j = 0; j < 32; ++j) {
    float s = b1[j];
    const float* w = w1 + j * 64;
#pragma unroll
    for (int c = 0; c < 64; ++c) s = fmaf(w[c], p[c], s);
    z[j] = fmaxf(s, 0.0f);
  }
#pragma unroll
  for (int o = 0; o < 2; ++o) {
    float s = b2[o];
    const float* w = w2 + o * 32;
#pragma unroll
    for (int j = 0; j < 32; ++j) s = fmaf(w[j], z[j], s);
    out[g * 2 + o] = s;
  }
}

// ---------------- driver ----------------
extern "C" void kernel_launch(void* const* d_in, const int* in_sizes, int n_in,
                              void* d_out, int out_size, void* d_ws, size_t ws_size,
                              hipStream_t stream) {
  (void)in_sizes; (void)n_in; (void)out_size; (void)ws_size;
  const float*     x     = (const float*)d_in[0];
  const long long* ei    = (const long long*)d_in[1];
  const long long* batch = (const long long*)d_in[2];
  const float* W1l = (const float*)d_in[3];
  const float* b1  = (const float*)d_in[4];
  const float* W1r = (const float*)d_in[5];
  const float* g1  = (const float*)d_in[6];
  const float* be1 = (const float*)d_in[7];
  const float* W2l = (const float*)d_in[8];
  const float* b2  = (const float*)d_in[9];
  const float* W2r = (const float*)d_in[10];
  const float* g2  = (const float*)d_in[11];
  const float* be2 = (const float*)d_in[12];
  const float* W3l = (const float*)d_in[13];
  const float* b3  = (const float*)d_in[14];
  const float* W3r = (const float*)d_in[15];
  const float* g3  = (const float*)d_in[16];
  const float* be3 = (const float*)d_in[17];
  const float* fc1w = (const float*)d_in[18];
  const float* fc1b = (const float*)d_in[19];
  const float* fc2w = (const float*)d_in[20];
  const float* fc2b = (const float*)d_in[21];
  float* out = (float*)d_out;

  char* ws = (char*)d_ws;
  size_t off = 0;
  auto walloc = [&](size_t bytes) { size_t p = off; off += (bytes + 255) & ~(size_t)255; return p; };
  float* INV   = (float*)(ws + walloc((size_t)N_NODES * 4));
  float* AGG   = (float*)(ws + walloc((size_t)N_NODES * 128 * 4));
  float* BUF1  = (float*)(ws + walloc((size_t)N_NODES * 128 * 4));
  float* BUF2  = (float*)(ws + walloc((size_t)N_NODES * 128 * 4));
  float* WLP   = (float*)(ws + walloc((size_t)128 * 128 * 4));   // padded Wl
  float* WRP   = (float*)(ws + walloc((size_t)128 * 128 * 4));   // padded Wr
  float* STATS = (float*)(ws + walloc(512 * 4));
  float* POOL  = (float*)(ws + walloc((size_t)(N_GRAPHS * 64 + N_GRAPHS) * 4));
  float* SUM = STATS;       float* SSQ = STATS + 128;
  float* SCL = STATS + 256; float* SHF = STATS + 384;
  float* PSUM = POOL;       float* PCNT = POOL + N_GRAPHS * 64;

  const long long* src = ei;
  const long long* dst = ei + N_EDGES;
  const float invN = 1.0f / (float)N_NODES;
  const int rowTiles = N_NODES / 16;            // 18750 (exact)

  // degrees -> 1/max(deg,1)
  k_zero<<<2048, 256, 0, stream>>>(INV, (long)N_NODES);
  k_deg<<<(N_EDGES + 255) / 256, 256, 0, stream>>>(dst, INV, N_EDGES);
  k_invdeg<<<(N_NODES + 255) / 256, 256, 0, stream>>>(INV, N_NODES);

  // ---------- layer 1: 18 -> 128 (lstride=32) ----------
  k_zero<<<2048, 256, 0, stream>>>(AGG, (long)N_NODES * 18);
  {
    long th = (long)N_EDGES << 5;   // cols padded to 32
    k_scatter<<<(unsigned)((th + 255) / 256), 256, 0, stream>>>(x, src, dst, AGG, N_EDGES, 18, 5);
  }
  k_padw<<<(128 * 32 + 255) / 256, 256, 0, stream>>>(W1l, WLP, 18, 128, 32, 5);
  k_padw<<<(128 * 32 + 255) / 256, 256, 0, stream>>>(W1r, WRP, 18, 128, 32, 5);
  k_sage_gemm<<<rowTiles, 256, 2 * 16 * 32 * 4, stream>>>(AGG, x, INV, WLP, WRP, b1, BUF1, 18, 128, 32, 5);
  k_zero<<<1, 256, 0, stream>>>(STATS, 256);
  k_colstats<<<(N_NODES + 255) / 256, 128, 0, stream>>>(BUF1, N_NODES, 128, 256, SUM, SSQ);
  k_bnfinalize<<<1, 128, 0, stream>>>(SUM, SSQ, g1, be1, SCL, SHF, 128, invN);
  k_bnrelu<<<(unsigned)(((long)N_NODES * 128 + 255) / 256), 256, 0, stream>>>(BUF1, SCL, SHF, (long)N_NODES * 128, 127);

  // ---------- layer 2: 128 -> 128 (lstride=128) ----------
  k_zero<<<2048, 256, 0, stream>>>(AGG, (long)N_NODES * 128);
  {
    long th = (long)N_EDGES << 5;   // 32 float4 chunks per edge
    k_scatter128<<<(unsigned)((th + 255) / 256), 256, 0, stream>>>(BUF1, src, dst, AGG, N_EDGES);
  }
  k_padw<<<(128 * 128 + 255) / 256, 256, 0, stream>>>(W2l, WLP, 128, 128, 128, 7);
  k_padw<<<(128 * 128 + 255) / 256, 256, 0, stream>>>(W2r, WRP, 128, 128, 128, 7);
  k_sage_gemm<<<rowTiles, 256, 2 * 16 * 128 * 4, stream>>>(AGG, BUF1, INV, WLP, WRP, b2, BUF2, 128, 128, 128, 7);
  k_zero<<<1, 256, 0, stream>>>(STATS, 256);
  k_colstats<<<(N_NODES + 255) / 256, 128, 0, stream>>>(BUF2, N_NODES, 128, 256, SUM, SSQ);
  k_bnfinalize<<<1, 128, 0, stream>>>(SUM, SSQ, g2, be2, SCL, SHF, 128, invN);
  k_bnrelu<<<(unsigned)(((long)N_NODES * 128 + 255) / 256), 256, 0, stream>>>(BUF2, SCL, SHF, (long)N_NODES * 128, 127);

  // ---------- layer 3: 128 -> 64 (lstride=128) ----------
  k_zero<<<2048, 256, 0, stream>>>(AGG, (long)N_NODES * 128);
  {
    long th = (long)N_EDGES << 5;
    k_scatter128<<<(unsigned)((th + 255) / 256), 256, 0, stream>>>(BUF2, src, dst, AGG, N_EDGES);
  }
  k_padw<<<(64 * 128 + 255) / 256, 256, 0, stream>>>(W3l, WLP, 128, 64, 128, 7);
  k_padw<<<(64 * 128 + 255) / 256, 256, 0, stream>>>(W3r, WRP, 128, 64, 128, 7);
  k_sage_gemm<<<rowTiles, 128, 2 * 16 * 128 * 4, stream>>>(AGG, BUF2, INV, WLP, WRP, b3, BUF1, 128, 64, 128, 7);
  k_zero<<<1, 256, 0, stream>>>(STATS, 256);
  k_colstats<<<(N_NODES + 255) / 256, 64, 0, stream>>>(BUF1, N_NODES, 64, 256, SUM, SSQ);
  k_bnfinalize<<<1, 64, 0, stream>>>(SUM, SSQ, g3, be3, SCL, SHF, 64, invN);
  k_bnrelu<<<(unsigned)(((long)N_NODES * 64 + 255) / 256), 256, 0, stream>>>(BUF1, SCL, SHF, (long)N_NODES * 64, 63);

  // ---------- pool + head ----------
  k_zero<<<(N_GRAPHS * 64 + N_GRAPHS + 255) / 256, 256, 0, stream>>>(POOL, (long)(N_GRAPHS * 64 + N_GRAPHS));
  k_pool<<<(unsigned)(((long)N_NODES * 64 + 255) / 256), 256, 0, stream>>>(BUF1, batch, PSUM, PCNT, N_NODES);
  k_head<<<2, 256, 0, stream>>>(PSUM, PCNT, fc1w, fc1b, fc2w, fc2b, out);
}